// AttentionPooling_70377334112806
// MI455X (gfx1250) — compile-verified
//
#include <hip/hip_runtime.h>
#include <math.h>

#define NUM_HEADS 4
#define KCHUNK 64
#define LDSTRIDE 68   // KCHUNK + 4 pad: A-fragment LDS reads become bank-conflict-free

typedef float v2f __attribute__((ext_vector_type(2)));
typedef float v8f __attribute__((ext_vector_type(8)));

// ---------------------------------------------------------------------------
// Float atomic max without CAS loop: positive floats order like ints,
// negative floats order inversely like uints. smax must be initialized to -inf.
// ---------------------------------------------------------------------------
__device__ __forceinline__ void atomic_max_float(float* addr, float val) {
  if (val >= 0.0f) {
    atomicMax((int*)addr, __float_as_int(val));
  } else {
    atomicMin((unsigned int*)addr, __float_as_uint(val));
  }
}

// ---------------------------------------------------------------------------
// Workspace init: zero two regions, fill smax with -inf.
// ---------------------------------------------------------------------------
__global__ void init_ws(float* z0, long n0, float* z1, long n1, float* smax, int ns) {
  long i = (long)blockIdx.x * blockDim.x + threadIdx.x;
  long stride = (long)gridDim.x * blockDim.x;
  for (long t = i; t < n0; t += stride) z0[t] = 0.0f;
  for (long t = i; t < n1; t += stride) z1[t] = 0.0f;
  for (long t = i; t < ns; t += stride) smax[t] = -INFINITY;
}

// ---------------------------------------------------------------------------
// Precompute q = query @ Wq.T + bq, then fold q through Wk:
//   s[h, c] = sum_d q[h*dh+d] * Wk[h*dh+d, c]
//   t[h]    = sum_d q[h*dh+d] * bk[h*dh+d]
// so scores[n,h] = (h_proj[n,:] . s[h,:] + t[h]) / sqrt(dh).
// One block of D threads. This eliminates the entire K projection GEMM.
// ---------------------------------------------------------------------------
__global__ void qst_kernel(const float* __restrict__ query, const float* __restrict__ in_w,
                           const float* __restrict__ in_b, int D,
                           float* __restrict__ sbuf, float* __restrict__ tbuf) {
  __shared__ float qs[1024];
  int j = threadIdx.x;
  int dh = D / NUM_HEADS;
  float acc = in_b[j];
  for (int c = 0; c < D; ++c) acc += query[c] * in_w[(long)j * D + c];
  qs[j] = acc;
  __syncthreads();
  for (int hh = 0; hh < NUM_HEADS; ++hh) {
    float s = 0.0f;
    const float* wk = in_w + (long)(D + hh * dh) * D;  // Wk rows for this head
    for (int d = 0; d < dh; ++d) s += qs[hh * dh + d] * wk[(long)d * D + j];
    sbuf[hh * D + j] = s;
  }
  if (j < NUM_HEADS) {
    float t = 0.0f;
    for (int d = 0; d < dh; ++d) t += qs[j * dh + d] * in_b[D + j * dh + d];
    tbuf[j] = t;
  }
}

// ---------------------------------------------------------------------------
// Scatter raw edge attrs: sumA[row[e], :] += edge_attr[e, :]; cnt[row[e]] += 1.
// segment_sum commutes with the edge linear layer, so we scatter the 64-wide
// raw attrs (204MB, L2-resident 12.8MB target) instead of 256-wide projections.
// ---------------------------------------------------------------------------
__global__ void edge_scatter(const float* __restrict__ eattr, const int* __restrict__ row,
                             float* __restrict__ sumA, float* __restrict__ cnt,
                             long E, int ED) {
  long total = E * (long)ED;
  long i = (long)blockIdx.x * blockDim.x + threadIdx.x;
  long stride = (long)gridDim.x * blockDim.x;
  for (long t = i; t < total; t += stride) {
    long e = t / ED;
    int c = (int)(t - e * ED);
    int r = row[e];
    atomicAdd(&sumA[(long)r * ED + c], eattr[t]);
    if (c == 0) atomicAdd(&cnt[r], 1.0f);
  }
}

// ===========================================================================
// WMMA GEMM core (fp32, V_WMMA_F32_16X16X4_F32).
// Block = 4 waves (128 thr). Block tile = 64 rows x 64 cols. Each wave owns a
// 16-col slice and FOUR 16-row accumulators, so one global B fragment feeds
// 4 WMMAs (4 WMMA : 5 loads). X is staged in 64-wide K chunks in LDS with a
// padded stride (68) -> conflict-free ds reads.
// fp32 fragment layout (ISA 7.12.2): A: lanes 0-15 hold (M=lane, K=k,k+1),
// lanes 16-31 hold (M=lane-16, K=k+2,k+3). B mirrors with N. C/D: VGPR r =
// rows r (lanes 0-15) and r+8 (lanes 16-31).
// ===========================================================================

// ---------------------------------------------------------------------------
// Fused node+edge projection:
//   h_proj[m,n] = h[m,:]@node_w[n,:].T + node_b[n]
//              + (sumA[m,:]/max(cnt,1))@edge_w[n,:].T + (cnt[m]>0)*edge_b[n]
// Virtual X = [h | mean_edge] with K = ND+ED; B switches source at col ND.
// ---------------------------------------------------------------------------
__global__ void hproj_wmma(const float* __restrict__ h, const float* __restrict__ sumA,
                           const float* __restrict__ cnt,
                           const float* __restrict__ node_w, const float* __restrict__ node_b,
                           const float* __restrict__ edge_w, const float* __restrict__ edge_b,
                           float* __restrict__ hproj, int N, int ND, int ED, int D) {
  const int K = ND + ED;                 // 192
  __shared__ float Xs[64 * LDSTRIDE];    // 17 KB
  int row0 = blockIdx.x * 64;
  int tid = threadIdx.x;
  int wave = tid >> 5, lane = tid & 31;
  int mrow = lane & 15;
  int n = blockIdx.y * 64 + wave * 16 + mrow;
  int khalf = (lane & 16) ? 2 : 0;

  v8f acc0 = {}, acc1 = {}, acc2 = {}, acc3 = {};

  for (int kc = 0; kc < K; kc += KCHUNK) {
    __syncthreads();
    for (int t = tid; t < 64 * KCHUNK; t += blockDim.x) {
      int r = t >> 6;          // / KCHUNK
      int c = t & (KCHUNK - 1);
      int m = row0 + r;
      int cg = kc + c;
      float val = 0.0f;
      if (m < N) {
        if (cg < ND) val = h[(long)m * ND + cg];
        else         val = sumA[(long)m * ED + (cg - ND)] / fmaxf(cnt[m], 1.0f);
      }
      Xs[r * LDSTRIDE + c] = val;
    }
    __syncthreads();
    for (int k = 0; k < KCHUNK; k += 4) {
      int kk = k + khalf;
      v2f b;
      if (kc + k < ND) b = *(const v2f*)&node_w[(long)n * ND + kc + kk];       // uniform
      else             b = *(const v2f*)&edge_w[(long)n * ED + (kc + kk - ND)];
      v2f a0 = *(const v2f*)&Xs[(mrow +  0) * LDSTRIDE + kk];
      v2f a1 = *(const v2f*)&Xs[(mrow + 16) * LDSTRIDE + kk];
      v2f a2 = *(const v2f*)&Xs[(mrow + 32) * LDSTRIDE + kk];
      v2f a3 = *(const v2f*)&Xs[(mrow + 48) * LDSTRIDE + kk];
      acc0 = __builtin_amdgcn_wmma_f32_16x16x4_f32(false, a0, false, b, (short)0, acc0, false, false);
      acc1 = __builtin_amdgcn_wmma_f32_16x16x4_f32(false, a1, false, b, (short)0, acc1, false, false);
      acc2 = __builtin_amdgcn_wmma_f32_16x16x4_f32(false, a2, false, b, (short)0, acc2, false, false);
      acc3 = __builtin_amdgcn_wmma_f32_16x16x4_f32(false, a3, false, b, (short)0, acc3, false, false);
    }
  }

  float bn = node_b[n];
  float be = edge_b[n];
  for (int r = 0; r < 8; ++r) {
    int mb = row0 + r + ((lane & 16) ? 8 : 0);
    v8f* accs[4] = {&acc0, &acc1, &acc2, &acc3};
    for (int mt = 0; mt < 4; ++mt) {
      int m = mb + mt * 16;
      if (m < N) {
        float mask = (cnt[m] > 0.0f) ? 1.0f : 0.0f;
        hproj[(long)m * D + n] = (*accs[mt])[r] + bn + mask * be;
      }
    }
  }
}

// ---------------------------------------------------------------------------
// Generic Y[M,Nout] = X[M,K] @ W[Nout,K].T + bias (V projection, output GEMM).
// ---------------------------------------------------------------------------
__global__ void gemm_xwT_wmma(const float* __restrict__ X, const float* __restrict__ W,
                              const float* __restrict__ bias, float* __restrict__ Y,
                              int M, int K, int Nout) {
  __shared__ float Xs[64 * LDSTRIDE];    // 17 KB
  int row0 = blockIdx.x * 64;
  int tid = threadIdx.x;
  int wave = tid >> 5, lane = tid & 31;
  int mrow = lane & 15;
  int n = blockIdx.y * 64 + wave * 16 + mrow;
  int khalf = (lane & 16) ? 2 : 0;

  v8f acc0 = {}, acc1 = {}, acc2 = {}, acc3 = {};

  for (int kc = 0; kc < K; kc += KCHUNK) {
    __syncthreads();
    for (int t = tid; t < 64 * KCHUNK; t += blockDim.x) {
      int r = t >> 6;
      int c = t & (KCHUNK - 1);
      int m = row0 + r;
      Xs[r * LDSTRIDE + c] = (m < M) ? X[(long)m * K + kc + c] : 0.0f;
    }
    __syncthreads();
    for (int k = 0; k < KCHUNK; k += 4) {
      int kk = k + khalf;
      v2f b = *(const v2f*)&W[(long)n * K + kc + kk];
      v2f a0 = *(const v2f*)&Xs[(mrow +  0) * LDSTRIDE + kk];
      v2f a1 = *(const v2f*)&Xs[(mrow + 16) * LDSTRIDE + kk];
      v2f a2 = *(const v2f*)&Xs[(mrow + 32) * LDSTRIDE + kk];
      v2f a3 = *(const v2f*)&Xs[(mrow + 48) * LDSTRIDE + kk];
      acc0 = __builtin_amdgcn_wmma_f32_16x16x4_f32(false, a0, false, b, (short)0, acc0, false, false);
      acc1 = __builtin_amdgcn_wmma_f32_16x16x4_f32(false, a1, false, b, (short)0, acc1, false, false);
      acc2 = __builtin_amdgcn_wmma_f32_16x16x4_f32(false, a2, false, b, (short)0, acc2, false, false);
      acc3 = __builtin_amdgcn_wmma_f32_16x16x4_f32(false, a3, false, b, (short)0, acc3, false, false);
    }
  }

  float bn = bias[n];
  for (int r = 0; r < 8; ++r) {
    int mb = row0 + r + ((lane & 16) ? 8 : 0);
    v8f* accs[4] = {&acc0, &acc1, &acc2, &acc3};
    for (int mt = 0; mt < 4; ++mt) {
      int m = mb + mt * 16;
      if (m < M) Y[(long)m * Nout + n] = (*accs[mt])[r] + bn;
    }
  }
}

// ---------------------------------------------------------------------------
// scores[n,h] = (h_proj[n,:] . s[h,:] + t[h]) / sqrt(dh); segment max into smax.
// ---------------------------------------------------------------------------
__global__ void scores_kernel(const float* __restrict__ hproj, const float* __restrict__ sbuf,
                              const float* __restrict__ tbuf, const int* __restrict__ batch,
                              float* __restrict__ smax, float* __restrict__ scores,
                              int N, int D) {
  int idx = blockIdx.x * blockDim.x + threadIdx.x;
  if (idx >= N * NUM_HEADS) return;
  int nI = idx >> 2;          // NUM_HEADS == 4
  int hh = idx & 3;
  const float* xr = hproj + (long)nI * D;
  const float* sr = sbuf + hh * D;
  float dot = tbuf[hh];
  for (int c = 0; c < D; c += 4) {
    float4 x = *(const float4*)&xr[c];
    float4 s = *(const float4*)&sr[c];
    dot += x.x * s.x + x.y * s.y + x.z * s.z + x.w * s.w;
  }
  float sc = dot * rsqrtf((float)(D / NUM_HEADS));
  scores[idx] = sc;
  atomic_max_float(&smax[batch[nI] * NUM_HEADS + hh], sc);
}

// ---------------------------------------------------------------------------
// w = exp(score - smax[g]); denom[g] += w.
// ---------------------------------------------------------------------------
__global__ void expw_kernel(const float* __restrict__ scores, const int* __restrict__ batch,
                            const float* __restrict__ smax, float* __restrict__ denom,
                            float* __restrict__ wexp, int N) {
  int idx = blockIdx.x * blockDim.x + threadIdx.x;
  if (idx >= N * NUM_HEADS) return;
  int nI = idx >> 2;
  int hh = idx & 3;
  int g = batch[nI];
  float w = __expf(scores[idx] - smax[g * NUM_HEADS + hh]);
  wexp[idx] = w;
  atomicAdd(&denom[g * NUM_HEADS + hh], w);
}

// ---------------------------------------------------------------------------
// pooled[g, c] += (w[n,h]/max(denom[g,h],1e-30)) * v[n, c], h = c / dh.
// ---------------------------------------------------------------------------
__global__ void pool_kernel(const float* __restrict__ wexp, const float* __restrict__ denom,
                            const int* __restrict__ batch, const float* __restrict__ v,
                            float* __restrict__ pooled, int N, int D) {
  long idx = (long)blockIdx.x * blockDim.x + threadIdx.x;
  long total = (long)N * D;
  if (idx >= total) return;
  int nI = (int)(idx / D);
  int c = (int)(idx - (long)nI * D);
  int hh = c / (D / NUM_HEADS);
  int g = batch[nI];
  float p = wexp[nI * NUM_HEADS + hh] / fmaxf(denom[g * NUM_HEADS + hh], 1e-30f);
  atomicAdd(&pooled[(long)g * D + c], p * v[idx]);
}

// ---------------------------------------------------------------------------
// Launch
// ---------------------------------------------------------------------------
extern "C" void kernel_launch(void* const* d_in, const int* in_sizes, int n_in,
                              void* d_out, int out_size, void* d_ws, size_t ws_size,
                              hipStream_t stream) {
  const float* h      = (const float*)d_in[0];
  const int*   eidx   = (const int*)d_in[1];   // [2,E], row = first E entries
  const float* eattr  = (const float*)d_in[2];
  const int*   batch  = (const int*)d_in[3];
  const float* node_w = (const float*)d_in[5];
  const float* node_b = (const float*)d_in[6];
  const float* edge_w = (const float*)d_in[7];
  const float* edge_b = (const float*)d_in[8];
  const float* query  = (const float*)d_in[9];
  const float* in_w   = (const float*)d_in[10];
  const float* in_b   = (const float*)d_in[11];
  const float* out_w  = (const float*)d_in[12];
  const float* out_b  = (const float*)d_in[13];
  float* out = (float*)d_out;

  int D  = in_sizes[9];            // query is (1, D)
  int ND = in_sizes[5] / D;        // node_w (D, ND)
  int ED = in_sizes[7] / D;        // edge_w (D, ED)
  int N  = in_sizes[0] / ND;       // h (N, ND)
  long E = in_sizes[1] / 2;        // edge_index (2, E)
  int G  = out_size / D;           // out (G, 1, D)

  // Workspace layout (floats), each region rounded to 16B alignment.
  float* ws = (float*)d_ws;
  long o = 0;
  auto take = [&](long nf) { float* p = ws + o; o += (nf + 3) & ~3L; return p; };
  float* sumA   = take((long)N * ED);
  float* cnt    = take(N);
  float* hproj  = take((long)N * D);
  float* vbuf   = take((long)N * D);
  float* scores = take((long)N * NUM_HEADS);
  float* wexp   = take((long)N * NUM_HEADS);
  float* sbuf   = take((long)NUM_HEADS * D);
  float* tbuf   = take(NUM_HEADS);
  float* smax   = take((long)G * NUM_HEADS);
  float* denom  = take((long)G * NUM_HEADS + (long)G * D);  // denom + pooled contiguous
  float* pooled = denom + G * NUM_HEADS;

  // 1) init: zero sumA+cnt (contiguous) and denom+pooled; smax = -inf
  {
    long z0 = (long)N * ED + N;
    long z1 = (long)G * NUM_HEADS + (long)G * D;
    long maxn = z0 > z1 ? z0 : z1;
    int blocks = (int)((maxn + 255) / 256);
    if (blocks > 262144) blocks = 262144;
    init_ws<<<blocks, 256, 0, stream>>>(sumA, z0, denom, z1, smax, G * NUM_HEADS);
  }
  // 2) q / s / t precompute
  qst_kernel<<<1, D, 0, stream>>>(query, in_w, in_b, D, sbuf, tbuf);
  // 3) edge scatter (raw attrs; projection folded into GEMM below)
  {
    long total = E * (long)ED;
    long blocks = (total + 255) / 256;
    if (blocks > 262144) blocks = 262144;
    edge_scatter<<<(int)blocks, 256, 0, stream>>>(eattr, eidx, sumA, cnt, E, ED);
  }
  // 4) fused node+edge projection (WMMA, K = ND+ED)
  {
    dim3 grid((N + 63) / 64, D / 64);
    hproj_wmma<<<grid, 128, 0, stream>>>(h, sumA, cnt, node_w, node_b, edge_w, edge_b,
                                         hproj, N, ND, ED, D);
  }
  // 5) scores + segment max; 6) exp + segment sum
  {
    int total = N * NUM_HEADS;
    scores_kernel<<<(total + 255) / 256, 256, 0, stream>>>(hproj, sbuf, tbuf, batch,
                                                           smax, scores, N, D);
    expw_kernel<<<(total + 255) / 256, 256, 0, stream>>>(scores, batch, smax, denom,
                                                         wexp, N);
  }
  // 7) V projection (WMMA)
  {
    dim3 grid((N + 63) / 64, D / 64);
    gemm_xwT_wmma<<<grid, 128, 0, stream>>>(hproj, in_w + 2L * D * D, in_b + 2 * D,
                                            vbuf, N, D, D);
  }
  // 8) softmax-weighted pooling scatter
  {
    long total = (long)N * D;
    pool_kernel<<<(int)((total + 255) / 256), 256, 0, stream>>>(wexp, denom, batch,
                                                                vbuf, pooled, N, D);
  }
  // 9) output GEMM (WMMA) -> d_out
  {
    dim3 grid((G + 63) / 64, D / 64);
    gemm_xwT_wmma<<<grid, 128, 0, stream>>>(pooled, out_w, out_b, out, G, D, D);
  }
}